// Net_56650618635135
// MI455X (gfx1250) — compile-verified
//
#include <hip/hip_runtime.h>

// v2f = A/B operand for V_WMMA_F32_16X16X4_F32 (2 VGPRs/lane), v8f = C/D (8 VGPRs/lane)
typedef __attribute__((ext_vector_type(2))) float v2f;
typedef __attribute__((ext_vector_type(8))) float v8f;

__global__ void zero_f32_kernel(float* __restrict__ p, int n) {
    int i = blockIdx.x * blockDim.x + threadIdx.x;
    if (i < n) p[i] = 0.0f;
}

// Each wave processes full tiles of 16 edges:
//   D(16x16) = A(16 edges x 16 ch) * B(Wa broadcast along N)
// via 4 chained V_WMMA_F32_16X16X4_F32 (K = 4 each, C seeded with inline 0),
// then one global_atomic_add_f32 per edge (bias added scalar-side).
// All loads/atomics are unconditional: only full tiles reach this kernel.
__global__ void edge_dot_scatter_wmma(const float* __restrict__ edge_attr,   // [E,16]
                                      const long long* __restrict__ dst,     // edge_index row 1, [E]
                                      const float* __restrict__ Wa,          // [16]
                                      const float* __restrict__ ba,          // [1]
                                      float* __restrict__ acc,               // [N] f32 accumulator (zeroed)
                                      long long nTiles) {                    // E_full / 16
    const int lane          = threadIdx.x & 31;
    const int half          = lane >> 4;          // 0: lanes 0-15, 1: lanes 16-31
    const int m             = lane & 15;          // edge-in-tile this lane feeds for A
    const int waveInBlock   = threadIdx.x >> 5;
    const int wavesPerBlock = blockDim.x >> 5;
    const long long waveId  = (long long)blockIdx.x * wavesPerBlock + waveInBlock;
    const long long nWaves  = (long long)gridDim.x * wavesPerBlock;

    const float bav = ba[0];

    // B operand: B[k][n] = Wa[k] for all n (broadcast along N).
    // Layout mirror of 32-bit A: k = 4*c + 2*half + vgpr_index.
    v2f b[4];
#pragma unroll
    for (int c = 0; c < 4; ++c) {
        b[c].x = Wa[4 * c + 2 * half + 0];
        b[c].y = Wa[4 * c + 2 * half + 1];
    }

    for (long long tile = waveId; tile < nTiles; tile += nWaves) {
        const long long e0 = tile * 16;

        // Prefetch the next tile this wave will touch (global_prefetch_b8 path).
        if (lane == 0) {
            const long long nt = tile + nWaves;
            if (nt < nTiles) __builtin_prefetch(edge_attr + nt * 256, 0, 1);
        }

        // A operand: lane supplies A[m][4c + 2*half + {0,1}] -> four unconditional
        // b64 loads; the wave consumes the full contiguous 1 KB tile (16 x 64 B).
        const float* row = edge_attr + (e0 + m) * 16 + 2 * half;
        v2f a[4];
#pragma unroll
        for (int c = 0; c < 4; ++c) {
            a[c] = *(const v2f*)(row + 4 * c);
        }

        // C = 0 (inline constant SRC2); bias is added after column extraction.
        v8f d = { 0.f, 0.f, 0.f, 0.f, 0.f, 0.f, 0.f, 0.f };
#pragma unroll
        for (int c = 0; c < 4; ++c) {
            // 8 args: (neg_a, A, neg_b, B, c_mod, C, reuse_a, reuse_b)
            d = __builtin_amdgcn_wmma_f32_16x16x4_f32(
                    false, a[c], false, b[c], (short)0, d, false, false);
        }

        // D layout: VGPR j, lanes 0-15 -> edge m=j; lanes 16-31 -> edge m=j+8,
        // identical across all 16 columns (B was broadcast). Hand one edge to
        // each of 16 lanes so the 16 atomics issue in parallel.
        const int sel = lane & 7;
        float val = 0.0f;
#pragma unroll
        for (int j = 0; j < 8; ++j) {
            if (sel == j) val = d[j];
        }
        const bool owner = (lane < 8) || (lane >= 16 && lane < 24);
        if (owner) {
            const long long e = e0 + (long long)(half * 8 + sel);
            atomicAdd(&acc[dst[e]], val + bav);   // global_atomic_add_f32
        }
    }
}

// Scalar tail for E % 16 edges (not launched when E is tile-aligned).
__global__ void edge_tail_kernel(const float* __restrict__ edge_attr,
                                 const long long* __restrict__ dst,
                                 const float* __restrict__ Wa,
                                 const float* __restrict__ ba,
                                 float* __restrict__ acc,
                                 long long eBase, int rem) {
    int i = blockIdx.x * blockDim.x + threadIdx.x;
    if (i < rem) {
        const long long e = eBase + i;
        const float* row = edge_attr + e * 16;
        float s = ba[0];
#pragma unroll
        for (int k = 0; k < 16; ++k) s = fmaf(row[k], Wa[k], s);
        atomicAdd(&acc[dst[e]], s);
    }
}

__global__ void finalize_kernel(const float* __restrict__ acc,
                                const float* __restrict__ Wd,
                                const float* __restrict__ bd,
                                float* __restrict__ out, int n) {
    int i = blockIdx.x * blockDim.x + threadIdx.x;
    if (i < n) out[i] = fmaf(acc[i], Wd[0], bd[0]);
}

extern "C" void kernel_launch(void* const* d_in, const int* in_sizes, int n_in,
                              void* d_out, int out_size, void* d_ws, size_t ws_size,
                              hipStream_t stream) {
    // setup_inputs order: x, edge_index, edge_attr, Wa, ba, Wd, bd
    const long long* edge_index = (const long long*)d_in[1];   // int64 [2, E]
    const float*     edge_attr  = (const float*)d_in[2];       // f32   [E, 16]
    const float*     Wa         = (const float*)d_in[3];       // f32   [16] (shape [1,16])
    const float*     ba         = (const float*)d_in[4];       // f32   [1]
    const float*     Wd         = (const float*)d_in[5];       // f32   [1]
    const float*     bd         = (const float*)d_in[6];       // f32   [1]

    const long long E      = (long long)in_sizes[2] / 16;      // 3.2M
    const long long nTiles = E >> 4;                           // full 16-edge tiles
    const int       rem    = (int)(E & 15);                    // 0 for this problem
    const int       N      = out_size;                         // 100K
    const long long* dst   = edge_index + E;                   // edge_index[1]

    float* acc = (float*)d_ws;                                 // N floats of scratch
    float* out = (float*)d_out;

    // 1) zero the per-node accumulator
    zero_f32_kernel<<<(N + 255) / 256, 256, 0, stream>>>(acc, N);

    // 2) WMMA edge-dot + scatter-add (grid-stride over 200K tiles of 16 edges)
    if (nTiles > 0) {
        edge_dot_scatter_wmma<<<1024, 256, 0, stream>>>(edge_attr, dst, Wa, ba, acc, nTiles);
    }
    if (rem > 0) {
        edge_tail_kernel<<<1, 64, 0, stream>>>(edge_attr, dst, Wa, ba, acc,
                                               nTiles * 16, rem);
    }

    // 3) out = acc * Wd + bd
    finalize_kernel<<<(N + 255) / 256, 256, 0, stream>>>(acc, Wd, bd, out, N);
}